// LGCNLayer_25984552141439
// MI455X (gfx1250) — compile-verified
//
#include <hip/hip_runtime.h>
#include <hip/hip_bf16.h>
#include <math.h>

// LightGCN propagation: out = Din^-1/2 * A * Dout^-1/2 * X
// N=50000 nodes, E=800000 edges, D=96 fp32 features.
// Entirely L2-resident on MI455X (x/out 19.2MB each vs 192MB L2).
// Strategy: device-built CSR (dst-binned) -> wave32-per-node gather with
// b96 row segments; edge-id batches staged through LDS via gfx1250 async
// global->LDS loads (ASYNCcnt) when the toolchain exposes the builtin.

#define N_FEAT 96

#if defined(__has_builtin)
#  if __has_builtin(__builtin_amdgcn_global_load_async_to_lds_b32)
#    define USE_ASYNC_LDS 1
#  endif
#endif
#ifndef USE_ASYNC_LDS
#  define USE_ASYNC_LDS 0
#endif

__device__ __forceinline__ void wait_asynccnt0() {
#if defined(__has_builtin)
#  if __has_builtin(__builtin_amdgcn_s_wait_asynccnt)
    __builtin_amdgcn_s_wait_asynccnt(0);
    return;
#  else
    asm volatile("s_wait_asynccnt 0" ::: "memory");
    return;
#  endif
#else
    asm volatile("s_wait_asynccnt 0" ::: "memory");
#endif
}

// 12-byte packet: one lane's contiguous 3-feature slice of a 96-wide row.
struct __align__(4) F3 { float a, b, c; };

// ---------------------------------------------------------------- utilities

__global__ __launch_bounds__(256) void zero_u32_kernel(unsigned* __restrict__ p, int n) {
    int i = blockIdx.x * blockDim.x + threadIdx.x;
    if (i < n) p[i] = 0u;
}

__global__ __launch_bounds__(256) void zero_f32_kernel(float* __restrict__ p, int n) {
    int i = blockIdx.x * blockDim.x + threadIdx.x;
    if (i < n) p[i] = 0.0f;
}

// ------------------------------------------------------- degree histograms

__global__ __launch_bounds__(256) void degree_kernel(const int* __restrict__ src,
                                                     const int* __restrict__ dst,
                                                     unsigned* __restrict__ deg_out,
                                                     unsigned* __restrict__ deg_in,
                                                     int nedges) {
    int stride = gridDim.x * blockDim.x;
    for (int e = blockIdx.x * blockDim.x + threadIdx.x; e < nedges; e += stride) {
        // Stream the index arrays ahead (gfx1250 global_prefetch_b8).
        __builtin_prefetch(&src[e + stride], 0, 1);
        __builtin_prefetch(&dst[e + stride], 0, 1);
        atomicAdd(&deg_out[src[e]], 1u);
        atomicAdd(&deg_in[dst[e]], 1u);
    }
}

// deg^{-1/2} with clip(deg,1): precise 1/sqrt (matches x**-0.5 closely)
__global__ __launch_bounds__(256) void scales_kernel(const unsigned* __restrict__ deg_out,
                                                     const unsigned* __restrict__ deg_in,
                                                     float* __restrict__ sc_out,
                                                     float* __restrict__ sc_in,
                                                     int nnodes) {
    int i = blockIdx.x * blockDim.x + threadIdx.x;
    if (i < nnodes) {
        unsigned dO = deg_out[i]; if (dO == 0u) dO = 1u;
        unsigned dI = deg_in[i];  if (dI == 0u) dI = 1u;
        sc_out[i] = 1.0f / sqrtf((float)dO);
        sc_in[i]  = 1.0f / sqrtf((float)dI);
    }
}

// xs = x * out_scale (row-wise), matching the reference's pre-scale
__global__ __launch_bounds__(256) void prescale_kernel(const float* __restrict__ x,
                                                       const float* __restrict__ sc_out,
                                                       float* __restrict__ xs,
                                                       int ntot) {
    int i = blockIdx.x * blockDim.x + threadIdx.x;
    if (i < ntot) xs[i] = x[i] * sc_out[i / N_FEAT];
}

// ------------------------------------------- exclusive scan (single block)

__global__ __launch_bounds__(1024) void scan_kernel(const unsigned* __restrict__ deg,
                                                    unsigned* __restrict__ row_ptr,
                                                    int n) {
    __shared__ unsigned s[1024];
    __shared__ unsigned carry_sh;
    int tid = threadIdx.x;
    if (tid == 0) carry_sh = 0u;
    __syncthreads();
    for (int base = 0; base < n; base += 1024) {
        int i = base + tid;
        unsigned v = (i < n) ? deg[i] : 0u;
        s[tid] = v;
        __syncthreads();
        for (int off = 1; off < 1024; off <<= 1) {
            unsigned t = (tid >= off) ? s[tid - off] : 0u;
            __syncthreads();
            s[tid] += t;
            __syncthreads();
        }
        unsigned incl  = s[tid];
        unsigned carry = carry_sh;
        if (i < n) row_ptr[i] = carry + incl - v;  // exclusive prefix
        __syncthreads();
        if (tid == 1023) carry_sh = carry + s[1023];
        __syncthreads();
    }
    if (tid == 0) row_ptr[n] = carry_sh;  // == total edge count
}

// ------------------------------------------------------ CSR bucket placement

__global__ __launch_bounds__(256) void bucket_kernel(const int* __restrict__ src,
                                                     const int* __restrict__ dst,
                                                     const unsigned* __restrict__ row_ptr,
                                                     unsigned* __restrict__ cursor,
                                                     unsigned* __restrict__ esrc,
                                                     int nedges) {
    int e = blockIdx.x * blockDim.x + threadIdx.x;
    if (e < nedges) {
        int d = dst[e];
        unsigned slot = atomicAdd(&cursor[d], 1u);
        esrc[row_ptr[d] + slot] = (unsigned)src[e];
    }
}

// ------------------------------------------- wave32-per-node gather (no fp atomics)
// Lane f owns features {3f, 3f+1, 3f+2}: one b96 load per edge, one b96 store.
// Edge-id batches are staged in LDS via async global->LDS loads (ASYNCcnt),
// then broadcast-read; fallback is a wave32 shuffle chain.

template <bool PRESCALED>
__global__ __launch_bounds__(256) void gather_kernel(const float* __restrict__ xsrc,
                                                     const unsigned* __restrict__ row_ptr,
                                                     const unsigned* __restrict__ esrc,
                                                     const float* __restrict__ sc_out,
                                                     const float* __restrict__ sc_in,
                                                     float* __restrict__ out,
                                                     int nnodes) {
#if USE_ASYNC_LDS
    __shared__ unsigned sbuf[256];           // 32 slots per wave, 8 waves/block
    const int wslot = (threadIdx.x >> 5) << 5;
#endif
    int gt   = blockIdx.x * blockDim.x + threadIdx.x;
    int node = gt >> 5;   // one wave32 per destination node
    int lane = gt & 31;
    if (node >= nnodes) return;

    int begin = (int)row_ptr[node];
    int end   = (int)row_ptr[node + 1];

    float a0 = 0.0f, a1 = 0.0f, a2 = 0.0f;

    for (int j0 = begin; j0 < end; j0 += 32) {
        int nb = end - j0; if (nb > 32) nb = 32;
#if USE_ASYNC_LDS
        if (lane < nb) {
            __builtin_amdgcn_global_load_async_to_lds_b32(
                (__attribute__((address_space(1))) int*)(esrc + j0 + lane),
                (__attribute__((address_space(3))) int*)(&sbuf[wslot + lane]),
                0, 0);
        }
        wait_asynccnt0();
#else
        int my = (lane < nb) ? (int)esrc[j0 + lane] : 0;  // cooperative batch load
#endif
        for (int t = 0; t < nb; ++t) {
#if USE_ASYNC_LDS
            int s = (int)sbuf[wslot + t];                  // LDS broadcast read
#else
            int s = __shfl(my, t, 32);                     // wave32 broadcast
#endif
            int su = __builtin_amdgcn_readfirstlane(s);    // uniform (SGPR) base
            const F3 v = *reinterpret_cast<const F3*>(xsrc + (size_t)su * N_FEAT + lane * 3);
            if (PRESCALED) {
                a0 += v.a; a1 += v.b; a2 += v.c;
            } else {
                float sc = sc_out[su];
                a0 = fmaf(v.a, sc, a0);
                a1 = fmaf(v.b, sc, a1);
                a2 = fmaf(v.c, sc, a2);
            }
        }
    }

    float si = sc_in[node];
    F3 w; w.a = a0 * si; w.b = a1 * si; w.c = a2 * si;
    *reinterpret_cast<F3*>(out + (size_t)node * N_FEAT + lane * 3) = w;  // b96 store
}

// ------------------------- fallback path (small workspace): fp32 atomics into out

__global__ __launch_bounds__(256) void edge_atomic_kernel(const int* __restrict__ src,
                                                          const int* __restrict__ dst,
                                                          const float* __restrict__ x,
                                                          const float* __restrict__ sc_out,
                                                          float* __restrict__ out,
                                                          int nedges) {
    int t = blockIdx.x * blockDim.x + threadIdx.x;
    int total = nedges * 24;          // 24 float4 chunks per 96-wide row
    if (t < total) {
        int e = t / 24;
        int c = t - e * 24;
        int sI = src[e];
        int dI = dst[e];
        float sc = sc_out[sI];
        const float4 v = *reinterpret_cast<const float4*>(x + (size_t)sI * N_FEAT + c * 4);
        float* o = out + (size_t)dI * N_FEAT + c * 4;
        atomicAdd(o + 0, v.x * sc);
        atomicAdd(o + 1, v.y * sc);
        atomicAdd(o + 2, v.z * sc);
        atomicAdd(o + 3, v.w * sc);
    }
}

__global__ __launch_bounds__(256) void post_scale_kernel(float* __restrict__ out,
                                                         const float* __restrict__ sc_in,
                                                         int ntot) {
    int i = blockIdx.x * blockDim.x + threadIdx.x;
    if (i < ntot) out[i] *= sc_in[i / N_FEAT];
}

// --------------------------------------------------------------- launcher

extern "C" void kernel_launch(void* const* d_in, const int* in_sizes, int n_in,
                              void* d_out, int out_size, void* d_ws, size_t ws_size,
                              hipStream_t stream) {
    const float* x   = (const float*)d_in[0];
    const int*   src = (const int*)d_in[1];
    const int*   dst = (const int*)d_in[2];
    float*       out = (float*)d_out;

    const int nedges = in_sizes[1];
    const int nnodes = in_sizes[0] / N_FEAT;
    const int ntot   = nnodes * N_FEAT;

    auto align256 = [](size_t v) { return (v + 255) & ~(size_t)255; };
    size_t off = 0;
    const size_t o_degout = off; off = align256(off + 4ull * nnodes);
    const size_t o_degin  = off; off = align256(off + 4ull * nnodes);
    const size_t o_scout  = off; off = align256(off + 4ull * nnodes);
    const size_t o_scin   = off; off = align256(off + 4ull * nnodes);
    const size_t o_rowptr = off; off = align256(off + 4ull * (nnodes + 1));
    const size_t o_cursor = off; off = align256(off + 4ull * nnodes);
    const size_t o_esrc   = off; off = align256(off + 4ull * nedges);
    const size_t need_csr = off;
    const size_t o_xs     = off; off = align256(off + 4ull * ntot);
    const size_t need_xs  = off;

    char* ws = (char*)d_ws;
    unsigned* deg_out = (unsigned*)(ws + o_degout);
    unsigned* deg_in  = (unsigned*)(ws + o_degin);
    float*    sc_out  = (float*)   (ws + o_scout);
    float*    sc_in   = (float*)   (ws + o_scin);
    unsigned* row_ptr = (unsigned*)(ws + o_rowptr);
    unsigned* cursor  = (unsigned*)(ws + o_cursor);
    unsigned* esrc    = (unsigned*)(ws + o_esrc);
    float*    xs      = (float*)   (ws + o_xs);

    const bool use_csr = (ws_size >= need_csr);
    const bool use_xs  = (ws_size >= need_xs);

    const dim3 blk(256);
    const int gN = (nnodes + 255) / 256;
    const int gE = (nedges + 255) / 256;
    const int gO = (ntot + 255) / 256;

    // degrees + scales (common to both paths)
    zero_u32_kernel<<<gN, blk, 0, stream>>>(deg_out, nnodes);
    zero_u32_kernel<<<gN, blk, 0, stream>>>(deg_in,  nnodes);
    degree_kernel<<<1024, blk, 0, stream>>>(src, dst, deg_out, deg_in, nedges);
    scales_kernel<<<gN, blk, 0, stream>>>(deg_out, deg_in, sc_out, sc_in, nnodes);

    if (use_csr) {
        zero_u32_kernel<<<gN, blk, 0, stream>>>(cursor, nnodes);
        scan_kernel<<<1, 1024, 0, stream>>>(deg_in, row_ptr, nnodes);
        bucket_kernel<<<gE, blk, 0, stream>>>(src, dst, row_ptr, cursor, esrc, nedges);
        const int gW = (nnodes * 32 + 255) / 256;  // one wave32 per node
        if (use_xs) {
            prescale_kernel<<<gO, blk, 0, stream>>>(x, sc_out, xs, ntot);
            gather_kernel<true><<<gW, blk, 0, stream>>>(xs, row_ptr, esrc, sc_out, sc_in, out, nnodes);
        } else {
            gather_kernel<false><<<gW, blk, 0, stream>>>(x, row_ptr, esrc, sc_out, sc_in, out, nnodes);
        }
    } else {
        zero_f32_kernel<<<gO, blk, 0, stream>>>(out, ntot);
        long long work = (long long)nedges * 24;
        edge_atomic_kernel<<<(int)((work + 255) / 256), blk, 0, stream>>>(src, dst, x, sc_out, out, nedges);
        post_scale_kernel<<<gO, blk, 0, stream>>>(out, sc_in, ntot);
    }
}